// GraphAttentionLayer_57999238365448
// MI455X (gfx1250) — compile-verified
//
#include <hip/hip_runtime.h>
#include <hip/hip_bf16.h>

typedef __attribute__((ext_vector_type(2))) float v2f;
typedef __attribute__((ext_vector_type(8))) float v8f;
typedef int v4i_vs __attribute__((vector_size(16)));   // b128 payload type for async-LDS builtin

#define NEG_SLOPE 0.2f

#ifndef __has_builtin
#define __has_builtin(x) 0
#endif
#if defined(__gfx1250__) && __has_builtin(__builtin_amdgcn_global_load_async_to_lds_b128)
#define USE_ASYNC_COPY 1
#else
#define USE_ASYNC_COPY 0
#endif

// ---- ordered-uint encoding for float atomic max (handles negatives) ----
__device__ __forceinline__ unsigned enc_f32(float f) {
    unsigned u = __float_as_uint(f);
    return (u & 0x80000000u) ? ~u : (u | 0x80000000u);
}
__device__ __forceinline__ float dec_f32(unsigned u) {
    unsigned b = (u & 0x80000000u) ? (u ^ 0x80000000u) : ~u;
    return __uint_as_float(b);
}

// ---- init: per-(node,head) running max = encoded(-inf) ----
__global__ void initm_kernel(unsigned* __restrict__ menc, int NH) {
    int idx = blockIdx.x * blockDim.x + threadIdx.x;
    if (idx < NH) menc[idx] = 0x007FFFFFu;   // enc(-inf) = ~0xFF800000
}

// ---- prep: Mpad[16][128] = a_src @ We (rows 4..15 zero); hconst[h] = be . a_src[h] ----
__global__ void prep_kernel(const float* __restrict__ att, const float* __restrict__ We,
                            const float* __restrict__ be,
                            float* __restrict__ Mpad, float* __restrict__ hconst) {
    for (int idx = threadIdx.x; idx < 16 * 128; idx += blockDim.x) {
        int hh = idx >> 7, k = idx & 127;
        float acc = 0.f;
        if (hh < 4) {
            for (int d = 0; d < 32; ++d) acc += att[hh * 64 + d] * We[d * 128 + k];
        }
        Mpad[idx] = acc;
    }
    if (threadIdx.x < 16) {
        float acc = 0.f;
        if (threadIdx.x < 4) {
            for (int d = 0; d < 32; ++d) acc += be[d] * att[threadIdx.x * 64 + d];
        }
        hconst[threadIdx.x] = acc;
    }
}

// ---- WMMA node GEMM: h[n,c] = sum_k x[n,k]*Ww[c,k] + Wb[c] ----
// block = 256 (8 waves); block covers 16 rows; wave w owns col tile [16w,16w+16)
__global__ void node_gemm_kernel(const float* __restrict__ x, const float* __restrict__ Ww,
                                 const float* __restrict__ Wb, float* __restrict__ hbuf,
                                 int N) {
    int wave = threadIdx.x >> 5;
    int lane = threadIdx.x & 31;
    int half = lane >> 4;
    int l15  = lane & 15;
    int row0 = blockIdx.x * 16;
    int col0 = wave * 16;

    int arow = row0 + l15; if (arow > N - 1) arow = N - 1;
    const float* xrow = x  + (size_t)arow * 128;
    const float* wrow = Ww + (size_t)(col0 + l15) * 128;  // B[k][col] = Ww[col][k]

    v8f c = {};
    #pragma unroll
    for (int k0 = 0; k0 < 128; k0 += 4) {
        v2f a, b;
        int ko = k0 + 2 * half;
        a.x = xrow[ko];     a.y = xrow[ko + 1];
        b.x = wrow[ko];     b.y = wrow[ko + 1];
        c = __builtin_amdgcn_wmma_f32_16x16x4_f32(false, a, false, b,
                                                  (short)0, c, false, false);
    }

    int col = col0 + l15;
    float bias = Wb[col];
    #pragma unroll
    for (int r = 0; r < 8; ++r) {
        int row = row0 + r + 8 * half;
        if (row < N) hbuf[(size_t)row * 128 + col] = c[r] + bias;
    }
}

// ---- per-(node,head) attention scores s_src, s_dst ----
__global__ void s_kernel(const float* __restrict__ hbuf, const float* __restrict__ att,
                         float* __restrict__ ssrc, float* __restrict__ sdst, int NH) {
    int idx = blockIdx.x * blockDim.x + threadIdx.x;
    if (idx >= NH) return;
    int head = idx & 3, n = idx >> 2;
    const float* hp = hbuf + (size_t)n * 128 + head * 32;
    const float* as = att + head * 64;
    const float* ad = as + 32;
    float acs = 0.f, acd = 0.f;
    #pragma unroll
    for (int d = 0; d < 32; ++d) { float hv = hp[d]; acs += hv * as[d]; acd += hv * ad[d]; }
    ssrc[idx] = acs; sdst[idx] = acd;
}

// ---- WMMA edge GEMM + logits + leaky relu + segment max ----
// block = 128 (4 waves); block tile = 64 edges x 128 K staged in LDS (async copy);
// each wave owns a 16-edge x 16-headcol WMMA tile (4 real head columns).
#define ETILE_ROWS 64
#define LDS_ROW_F  132   // 128 + 4 pad floats -> bank-conflict-free ds_load_b64

__global__ void edge_gemm_kernel(const float* __restrict__ ea, const float* __restrict__ Mpad,
                                 const float* __restrict__ hconst, const int* __restrict__ ei,
                                 const float* __restrict__ ssrc, const float* __restrict__ sdst,
                                 float* __restrict__ logits, unsigned* __restrict__ menc,
                                 int E) {
    __shared__ float atile[ETILE_ROWS * LDS_ROW_F];

    int tid  = threadIdx.x;
    int wave = tid >> 5;
    int lane = tid & 31;
    int half = lane >> 4;
    int l15  = lane & 15;
    long long e0b = (long long)blockIdx.x * ETILE_ROWS;

    // ---- stage 64x128 f32 tile of edge_attr into LDS, fully coalesced b128 ----
    size_t gmax16 = (size_t)E * 512 - 16;
    #pragma unroll
    for (int r = 0; r < 16; ++r) {
        int chunk = r * 128 + tid;            // 2048 chunks of 16B
        int row   = chunk >> 5;
        int c16   = chunk & 31;
        size_t goff = ((size_t)e0b + row) * 512 + (size_t)c16 * 16;
        if (goff > gmax16) goff = gmax16;     // clamp (results for OOB rows discarded)
        size_t loff = (size_t)row * (LDS_ROW_F * 4) + (size_t)c16 * 16;
#if USE_ASYNC_COPY
        __builtin_amdgcn_global_load_async_to_lds_b128(
            (__attribute__((address_space(1))) v4i_vs*)((const char*)ea + goff),
            (__attribute__((address_space(3))) v4i_vs*)((char*)atile + loff),
            0, 0);
#else
        *(float4*)((char*)atile + loff) = *(const float4*)((const char*)ea + goff);
#endif
    }
#if USE_ASYNC_COPY
    asm volatile("s_wait_asynccnt 0x0" ::: "memory");
#endif
    __syncthreads();

    // ---- WMMA from LDS (A) and global (B = Mpad^T, L1-resident) ----
    const float* lrow = atile + (size_t)(wave * 16 + l15) * LDS_ROW_F;
    const float* mrow = Mpad  + (size_t)l15 * 128;
    v8f c = {};
    #pragma unroll
    for (int k0 = 0; k0 < 128; k0 += 4) {
        int ko = k0 + 2 * half;
        v2f a, b;
        a.x = lrow[ko]; a.y = lrow[ko + 1];
        b.x = mrow[ko]; b.y = mrow[ko + 1];
        c = __builtin_amdgcn_wmma_f32_16x16x4_f32(false, a, false, b,
                                                  (short)0, c, false, false);
    }

    long long e0 = e0b + wave * 16;
    int head = l15;
    if (head < 4) {
        float hc = hconst[head];
        #pragma unroll
        for (int r = 0; r < 8; ++r) {
            long long e = e0 + r + 8 * half;
            if (e < E) {
                int src = ei[e];
                int dst = ei[(long long)E + e];
                float lg = c[r] + hc + ssrc[src * 4 + head] + sdst[dst * 4 + head];
                lg = (lg > 0.f) ? lg : NEG_SLOPE * lg;
                logits[e * 4 + head] = lg;
                atomicMax(&menc[dst * 4 + head], enc_f32(lg));
            }
        }
    }
}

// ---- decode max; non-finite -> 0 (in place: unsigned -> float) ----
__global__ void mfinal_kernel(unsigned* __restrict__ menc, int NH) {
    int idx = blockIdx.x * blockDim.x + threadIdx.x;
    if (idx >= NH) return;
    float f = dec_f32(menc[idx]);
    if (!isfinite(f)) f = 0.f;
    ((float*)menc)[idx] = f;
}

// ---- ex = exp(logit - m[dst]); denom[dst] += ex (logits overwritten with ex) ----
__global__ void exp_kernel(float* __restrict__ logits, const float* __restrict__ mfl,
                           const int* __restrict__ ei, float* __restrict__ denom, int E) {
    int idx = blockIdx.x * blockDim.x + threadIdx.x;
    if (idx >= E * 4) return;
    int e = idx >> 2, head = idx & 3;
    int dst = ei[(long long)E + e];
    float ex = __expf(logits[idx] - mfl[dst * 4 + head]);
    logits[idx] = ex;
    atomicAdd(&denom[dst * 4 + head], ex);
}

// ---- out[dst] += alpha * h[src]; one wave per edge, lane = d ----
__global__ void scatter_kernel(const float* __restrict__ hbuf, const float* __restrict__ exv,
                               const float* __restrict__ denom, const int* __restrict__ ei,
                               float* __restrict__ out, int E) {
    int wave = threadIdx.x >> 5;
    int lane = threadIdx.x & 31;
    long long e = (long long)blockIdx.x * 8 + wave;
    if (e >= E) return;
    int src = ei[e];
    int dst = ei[(long long)E + e];
    const float* hs = hbuf + (size_t)src * 128;
    float* op = out + (size_t)dst * 128;
    #pragma unroll
    for (int head = 0; head < 4; ++head) {
        float alpha = exv[e * 4 + head] / (denom[dst * 4 + head] + 1e-16f);
        atomicAdd(&op[head * 32 + lane], alpha * hs[head * 32 + lane]);
    }
}

static inline size_t align_up(size_t v) { return (v + 255) & ~(size_t)255; }

extern "C" void kernel_launch(void* const* d_in, const int* in_sizes, int n_in,
                              void* d_out, int out_size, void* d_ws, size_t ws_size,
                              hipStream_t stream) {
    const float* x   = (const float*)d_in[0];
    const int*   ei  = (const int*)  d_in[1];   // [2,E]: src row then dst row
    const float* ea  = (const float*)d_in[2];
    const float* Ww  = (const float*)d_in[3];
    const float* Wb  = (const float*)d_in[4];
    const float* att = (const float*)d_in[5];
    const float* We  = (const float*)d_in[6];
    const float* be  = (const float*)d_in[7];

    const int N = in_sizes[0] / 128;
    const int E = in_sizes[1] / 2;
    const int NH = N * 4;

    // workspace carve-up
    char* ws = (char*)d_ws;
    size_t off = 0;
    float*    hbuf   = (float*)(ws + off); off = align_up(off + (size_t)N * 128 * 4);
    float*    ssrc   = (float*)(ws + off); off = align_up(off + (size_t)NH * 4);
    float*    sdst   = (float*)(ws + off); off = align_up(off + (size_t)NH * 4);
    unsigned* menc   = (unsigned*)(ws + off); off = align_up(off + (size_t)NH * 4);
    float*    denom  = (float*)(ws + off); off = align_up(off + (size_t)NH * 4);
    float*    logits = (float*)(ws + off); off = align_up(off + (size_t)E * 4 * 4);
    float*    Mpad   = (float*)(ws + off); off = align_up(off + (size_t)16 * 128 * 4);
    float*    hconst = (float*)(ws + off); off = align_up(off + 16 * 4);
    (void)ws_size;

    // 0) zero accumulators (graph-capture-safe memset nodes)
    (void)hipMemsetAsync(d_out, 0, (size_t)N * 128 * sizeof(float), stream);
    (void)hipMemsetAsync(denom, 0, (size_t)NH * sizeof(float), stream);
    initm_kernel<<<(NH + 255) / 256, 256, 0, stream>>>(menc, NH);
    prep_kernel<<<1, 256, 0, stream>>>(att, We, be, Mpad, hconst);

    // 1) node GEMM (WMMA f32 16x16x4)
    node_gemm_kernel<<<(N + 15) / 16, 256, 0, stream>>>(x, Ww, Wb, hbuf, N);

    // 2) per-node attention scores
    s_kernel<<<(NH + 255) / 256, 256, 0, stream>>>(hbuf, att, ssrc, sdst, NH);

    // 3) edge GEMM + logits + segment max (async LDS staging + WMMA)
    edge_gemm_kernel<<<(E + ETILE_ROWS - 1) / ETILE_ROWS, 128, 0, stream>>>(
        ea, Mpad, hconst, ei, ssrc, sdst, logits, menc, E);

    // 4) finalize max
    mfinal_kernel<<<(NH + 255) / 256, 256, 0, stream>>>(menc, NH);

    // 5) exp + denom
    exp_kernel<<<(E * 4 + 255) / 256, 256, 0, stream>>>(logits, (const float*)menc, ei, denom, E);

    // 6) weighted scatter
    scatter_kernel<<<(E + 7) / 8, 256, 0, stream>>>(hbuf, logits, denom, ei, (float*)d_out, E);
}